// GCF_57475252355458
// MI455X (gfx1250) — compile-verified
//
#include <hip/hip_runtime.h>
#include <math.h>

#define N_USER 60000
#define N_ITEM 40000
#define NN     (N_USER + N_ITEM)      // 100000 nodes
#define D      64
#define NNZ_E  1600000
#define BB     16384
#define ND     ((size_t)NN * D)       // 6,400,000

typedef __attribute__((ext_vector_type(2))) float v2f;
typedef __attribute__((ext_vector_type(8))) float v8f;

__device__ __forceinline__ void atomAddF(float* p, float v) {
    __hip_atomic_fetch_add(p, v, __ATOMIC_RELAXED, __HIP_MEMORY_SCOPE_AGENT);
}

// ---------------------------------------------------------------- init: X = [uEmbd; iEmbd], finalAcc = X
__global__ void k_init(const float* __restrict__ uE, const float* __restrict__ iE,
                       float* __restrict__ X, float* __restrict__ finalAcc) {
    size_t i = (size_t)blockIdx.x * blockDim.x + threadIdx.x;
    if (i >= ND) return;
    float v = (i < (size_t)N_USER * D) ? uE[i] : iE[i - (size_t)N_USER * D];
    X[i] = v;
    finalAcc[i] = v;
}

// ---------------------------------------------------------------- zero SpMM accumulators + this relation's att scalar
__global__ void k_zero(float* __restrict__ S1, float* __restrict__ S2, float* __restrict__ slot) {
    size_t i = (size_t)blockIdx.x * blockDim.x + threadIdx.x;
    if (i == 0) *slot = 0.f;
    if (i < ND) { S1[i] = 0.f; S2[i] = 0.f; }
}

// ---------------------------------------------------------------- fused SpMM: S1 += val*X[col], S2 += val*X[col]^2 (scatter to row)
__global__ void k_spmm(const int* __restrict__ row, const int* __restrict__ col,
                       const float* __restrict__ val, const float* __restrict__ X,
                       float* __restrict__ S1, float* __restrict__ S2) {
    size_t gid = (size_t)blockIdx.x * blockDim.x + threadIdx.x;
    size_t e = gid >> 6;
    if (e >= (size_t)NNZ_E) return;
    int d = (int)(gid & 63);
    int r = row[e], c = col[e];
    float v = val[e];
    float x = X[(size_t)c * D + d];
    atomAddF(&S1[(size_t)r * D + d], v * x);
    atomAddF(&S2[(size_t)r * D + d], v * x * x);
}

// ---------------------------------------------------------------- dense phase (one wave per 16-row tile):
// hsum = (S1+X)@WL + S2@WI + (bL+bI)  via f32 WMMA;  att += sum(tanh(hsum@aW+ab)@avec)
__global__ __launch_bounds__(32)
void k_dense(const float* __restrict__ S1, const float* __restrict__ S2,
             const float* __restrict__ X,  float* __restrict__ H,
             const float* __restrict__ WL, const float* __restrict__ WI,
             const float* __restrict__ bL, const float* __restrict__ bI,
             const float* __restrict__ aW, const float* __restrict__ ab,
             const float* __restrict__ avec, float* __restrict__ attsum) {
    __shared__ float lds[16 * 64];
    const int lane = threadIdx.x;
    const int half = lane >> 4;       // 0: K even pair, 1: K odd pair (A layout)
    const int lm   = lane & 15;
    const int m0   = blockIdx.x * 16;
    const size_t rowOff = (size_t)(m0 + lm) * D;

    v8f acc[4] = {};                  // 4 n-tiles of 16 cols each

    for (int kk = 0; kk < 16; ++kk) { // K = 64, 4 per WMMA
        const int k0 = kk * 4 + half * 2;
        v2f a1, a2;
        a1.x = S1[rowOff + k0]     + X[rowOff + k0];
        a1.y = S1[rowOff + k0 + 1] + X[rowOff + k0 + 1];
        a2.x = S2[rowOff + k0];
        a2.y = S2[rowOff + k0 + 1];
#pragma unroll
        for (int nt = 0; nt < 4; ++nt) {
            const int colN = nt * 16 + lm;
            v2f bl, bi;
            bl.x = WL[(size_t)k0 * D + colN];
            bl.y = WL[(size_t)(k0 + 1) * D + colN];
            bi.x = WI[(size_t)k0 * D + colN];
            bi.y = WI[(size_t)(k0 + 1) * D + colN];
            acc[nt] = __builtin_amdgcn_wmma_f32_16x16x4_f32(
                false, a1, false, bl, (short)0, acc[nt], false, false);
            acc[nt] = __builtin_amdgcn_wmma_f32_16x16x4_f32(
                false, a2, false, bi, (short)0, acc[nt], false, false);
        }
    }

    // bias, store hsum to global H and LDS (C layout: VGPR v -> M=v+8*half, N=nt*16+lm)
#pragma unroll
    for (int nt = 0; nt < 4; ++nt) {
        const int colN = nt * 16 + lm;
        const float bias = bL[colN] + bI[colN];
#pragma unroll
        for (int v = 0; v < 8; ++v) {
            const int m = v + half * 8;
            const float h = acc[nt][v] + bias;
            H[(size_t)(m0 + m) * D + colN] = h;
            lds[m * 64 + colN] = h;
        }
    }
    __syncthreads();

    // attention contribution of this tile: sum_i sum_j tanh(sum_k hsum[i][k]*aW[k][j] + ab[j]) * avec[j]
    float part = 0.f;
    for (int t = 0; t < 32; ++t) {
        const int idx = lane + t * 32;       // 0..1023 over (i,j)
        const int i = idx >> 6, j = idx & 63;
        float s = ab[j];
        for (int k = 0; k < 64; ++k) s += lds[i * 64 + k] * aW[(size_t)k * D + j];
        part += tanhf(s) * avec[j];
    }
    for (int off = 16; off > 0; off >>= 1) part += __shfl_down(part, off, 32);
    if (lane == 0) atomAddF(attsum, part);
}

// ---------------------------------------------------------------- beta = softmax([main,add,trust]); feats = sum beta_r * H_r; finalAcc += feats
__global__ void k_combine(const float* __restrict__ Hm, const float* __restrict__ Ha,
                          const float* __restrict__ Ht, float* __restrict__ X,
                          float* __restrict__ finalAcc, const float* __restrict__ attsum) {
    size_t i = (size_t)blockIdx.x * blockDim.x + threadIdx.x;
    if (i >= ND) return;
    const float invN = 1.0f / (float)NN;
    float w0 = attsum[0] * invN;   // main
    float w1 = attsum[1] * invN;   // add
    float w2 = attsum[2] * invN;   // trust
    float mx = fmaxf(w0, fmaxf(w1, w2));
    float e0 = expf(w0 - mx), e1 = expf(w1 - mx), e2 = expf(w2 - mx);
    float inv = 1.0f / (e0 + e1 + e2);
    float f = (e0 * Hm[i] + e1 * Ha[i] + e2 * Ht[i]) * inv;
    X[i] = f;
    finalAcc[i] += f;
}

// ---------------------------------------------------------------- final = finalAcc / 3
__global__ void k_scale(float* __restrict__ finalOut) {
    size_t i = (size_t)blockIdx.x * blockDim.x + threadIdx.x;
    if (i >= ND) return;
    finalOut[i] *= (1.0f / 3.0f);
}

// ---------------------------------------------------------------- gather userE/itemE + 3-layer MLP (one wave per pair)
__global__ __launch_bounds__(32)
void k_mlp(const int* __restrict__ userIdx, const int* __restrict__ itemIdx,
           const float* __restrict__ finalE,
           const float* __restrict__ W1, const float* __restrict__ b1,
           const float* __restrict__ W2, const float* __restrict__ b2,
           const float* __restrict__ W3, const float* __restrict__ b3,
           float* __restrict__ pred, float* __restrict__ outU, float* __restrict__ outI) {
    __shared__ float h[128];
    __shared__ float hh[64];
    const int b = blockIdx.x, lane = threadIdx.x;
    const int u  = userIdx[b];
    const int it = itemIdx[b] + N_USER;
    for (int d = lane; d < 64; d += 32) {
        float uv = finalE[(size_t)u * D + d];
        float iv = finalE[(size_t)it * D + d];
        h[d] = uv; h[64 + d] = iv;
        outU[(size_t)b * 64 + d] = uv;
        outI[(size_t)b * 64 + d] = iv;
    }
    __syncthreads();
    for (int j = lane; j < 64; j += 32) {
        float s = b1[j];
        for (int k = 0; k < 128; ++k) s += h[k] * W1[k * 64 + j];
        hh[j] = fmaxf(s, 0.f);
    }
    __syncthreads();
    float s2 = b2[lane];
    for (int k = 0; k < 64; ++k) s2 += hh[k] * W2[k * 32 + lane];
    float p = s2 * W3[lane];
    for (int off = 16; off > 0; off >>= 1) p += __shfl_down(p, off, 32);
    if (lane == 0) pred[b] = p + b3[0];
}

// ----------------------------------------------------------------
extern "C" void kernel_launch(void* const* d_in, const int* in_sizes, int n_in,
                              void* d_out, int out_size, void* d_ws, size_t ws_size,
                              hipStream_t stream) {
    const int*   userIdx = (const int*)d_in[0];
    const int*   itemIdx = (const int*)d_in[1];
    const float* uEmbd   = (const float*)d_in[2];
    const float* iEmbd   = (const float*)d_in[3];
    // relations in processing order: 0=lap(main), 1=trust, 2=add
    const int*   rrow[3] = { (const int*)d_in[4], (const int*)d_in[7], (const int*)d_in[10] };
    const int*   rcol[3] = { (const int*)d_in[5], (const int*)d_in[8], (const int*)d_in[11] };
    const float* rval[3] = { (const float*)d_in[6], (const float*)d_in[9], (const float*)d_in[12] };
    const float* lin_W   = (const float*)d_in[13];
    const float* inter_W = (const float*)d_in[14];
    // per relation: main->attM, trust->attT, add->attA
    const float* attW[3] = { (const float*)d_in[15], (const float*)d_in[17], (const float*)d_in[16] };
    const float* lin_b   = (const float*)d_in[18];
    const float* inter_b = (const float*)d_in[19];
    const float* attB[3] = { (const float*)d_in[20], (const float*)d_in[22], (const float*)d_in[21] };
    const float* avec[3] = { (const float*)d_in[23], (const float*)d_in[25], (const float*)d_in[24] };
    const float* W1 = (const float*)d_in[26];
    const float* b1 = (const float*)d_in[27];
    const float* W2 = (const float*)d_in[28];
    const float* b2 = (const float*)d_in[29];
    const float* W3 = (const float*)d_in[30];
    const float* b3 = (const float*)d_in[31];

    float* ws = (float*)d_ws;
    float* X      = ws;
    float* S1     = ws + ND;
    float* S2     = ws + 2 * ND;
    float* Hmain  = ws + 3 * ND;
    float* Hadd   = ws + 4 * ND;
    float* Htrust = ws + 5 * ND;
    float* attS   = ws + 6 * ND;       // [0]=main [1]=add [2]=trust
    float* relH[3]  = { Hmain, Htrust, Hadd };   // processing order
    int    relSlot[3] = { 0, 2, 1 };

    float* pred = (float*)d_out;
    float* outU = pred + BB;
    float* outI = outU + (size_t)BB * 64;
    float* outF = outI + (size_t)BB * 64;        // also serves as the 3-stage mean accumulator

    const int T = 256;
    const unsigned gND   = (unsigned)((ND + T - 1) / T);
    const unsigned gSPMM = (unsigned)(((size_t)NNZ_E * 64 + T - 1) / T);
    const unsigned gTILE = NN / 16;              // 6250

    k_init<<<gND, T, 0, stream>>>(uEmbd, iEmbd, X, outF);

    for (int l = 0; l < 2; ++l) {
        const float* WL = lin_W   + (size_t)l * D * D;
        const float* WI = inter_W + (size_t)l * D * D;
        const float* bL = lin_b   + (size_t)l * D;
        const float* bI = inter_b + (size_t)l * D;
        for (int r = 0; r < 3; ++r) {
            k_zero<<<gND, T, 0, stream>>>(S1, S2, attS + relSlot[r]);
            k_spmm<<<gSPMM, T, 0, stream>>>(rrow[r], rcol[r], rval[r], X, S1, S2);
            k_dense<<<gTILE, 32, 0, stream>>>(S1, S2, X, relH[r], WL, WI, bL, bI,
                                              attW[r] + (size_t)l * D * D,
                                              attB[r] + (size_t)l * D,
                                              avec[r] + (size_t)l * D,
                                              attS + relSlot[r]);
        }
        k_combine<<<gND, T, 0, stream>>>(Hmain, Hadd, Htrust, X, outF, attS);
    }

    k_scale<<<gND, T, 0, stream>>>(outF);
    k_mlp<<<BB, 32, 0, stream>>>(userIdx, itemIdx, outF, W1, b1, W2, b2, W3, b3,
                                 pred, outU, outI);
}